// Dempster_Shafer_module_18021682775041
// MI455X (gfx1250) — compile-verified
//
#include <hip/hip_runtime.h>
#include <hip/hip_bf16.h>

typedef __attribute__((ext_vector_type(2))) float v2f;
typedef __attribute__((ext_vector_type(4))) float v4f;
typedef __attribute__((ext_vector_type(8))) float v8f;

#define EPS_F 1e-4f

constexpr int B_TOTAL = 8192;
constexpr int F_DIM   = 512;
constexpr int P_DIM   = 64;
constexpr int C_DIM   = 100;
constexpr int ROWS_PER_BLOCK = 64;
constexpr int NTHREADS = 64;
constexpr int S_PITCH = ROWS_PER_BLOCK + 4;   // pad to dodge 64-bank conflicts

__global__ __launch_bounds__(NTHREADS) void ds_fused(
    const float* __restrict__ x, const float* __restrict__ w,
    const float* __restrict__ xi, const float* __restrict__ eta,
    const float* __restrict__ beta, float* __restrict__ out)
{
  __shared__ float u_lds[P_DIM * C_DIM];      // 25.6 KB : u = beta^2 / sum
  __shared__ float s_lds[P_DIM * S_PITCH];    // 17.4 KB : unnormalized si, [p][row]
  __shared__ float xx_lds[ROWS_PER_BLOCK];
  __shared__ float ww_lds[P_DIM];
  __shared__ float al_lds[P_DIM];
  __shared__ float ga_lds[P_DIM];

  const int tid  = threadIdx.x;
  const int lane = tid & 31;
  const int wave = tid >> 5;
  const int half = lane >> 4;                 // 0: lanes 0-15, 1: lanes 16-31
  const int l15  = lane & 15;
  const int row_base = blockIdx.x * ROWS_PER_BLOCK;

  // ---------------- prologue: per-prototype constants (thread p = tid) -------
  {
    const int p = tid;
    const float* wr = w + p * F_DIM;
    float acc = 0.f;
    #pragma unroll 4
    for (int k = 0; k < F_DIM; k += 4) {
      v4f q = *(const v4f*)(wr + k);
      acc += q.x*q.x + q.y*q.y + q.z*q.z + q.w*q.w;
    }
    ww_lds[p] = acc;                          // ||w_p||^2
    float e = eta[p];
    ga_lds[p] = e * e;                        // gamma
    al_lds[p] = 1.f / (1.f + __expf(-xi[p])); // alpha = sigmoid(xi)
    const float* br = beta + p * C_DIM;
    float bs = 0.f;
    #pragma unroll 4
    for (int c = 0; c < C_DIM; ++c) { float b = br[c]; bs += b * b; }
    float ibs = 1.f / bs;
    #pragma unroll 4
    for (int c = 0; c < C_DIM; ++c) { float b = br[c]; u_lds[p*C_DIM + c] = b*b*ibs; }
  }

  // ---------------- GEMM: x @ w^T via V_WMMA_F32_16X16X4_F32 ----------------
  // A frag (16x4 f32): lane l15 = row, K = i + 2*half  -> float2 load
  // B frag (4x16 f32): lane l15 = col, K = i + 2*half  -> float2 load of w row
  const int koff = half * 2;
  const float* xr0 = x + (size_t)(row_base + wave*32 + l15) * F_DIM + koff;
  const float* xr1 = xr0 + 16 * F_DIM;                 // second M-tile
  const float* wrb = w + (size_t)l15 * F_DIM + koff;   // N-tile stride = 16*F_DIM

  v8f acc[2][4] = {};
  float xx0 = 0.f, xx1 = 0.f;

  #pragma unroll 4
  for (int k = 0; k < F_DIM; k += 4) {
    v2f a0 = *(const v2f*)(xr0 + k);
    v2f a1 = *(const v2f*)(xr1 + k);
    xx0 += a0.x*a0.x + a0.y*a0.y;                      // ||x||^2 partial
    xx1 += a1.x*a1.x + a1.y*a1.y;
    #pragma unroll
    for (int t = 0; t < 4; ++t) {
      v2f b = *(const v2f*)(wrb + t * 16 * F_DIM + k);
      acc[0][t] = __builtin_amdgcn_wmma_f32_16x16x4_f32(
          false, a0, false, b, (short)0, acc[0][t], false, false);
      acc[1][t] = __builtin_amdgcn_wmma_f32_16x16x4_f32(
          false, a1, false, b, (short)0, acc[1][t], false, false);
    }
  }

  // combine ||x||^2 halves (K 0..1 in lanes<16, K 2..3 in lanes>=16)
  xx0 += __shfl_xor(xx0, 16, 32);
  xx1 += __shfl_xor(xx1, 16, 32);
  if (half == 0) {
    xx_lds[wave*32 + l15]      = xx0;
    xx_lds[wave*32 + 16 + l15] = xx1;
  }
  __syncthreads();

  // epilogue: d = xx + ww - 2*xw ; s = exp(-gamma*d)*alpha ; store [p][row]
  #pragma unroll
  for (int mt = 0; mt < 2; ++mt) {
    #pragma unroll
    for (int t = 0; t < 4; ++t) {
      #pragma unroll
      for (int j = 0; j < 8; ++j) {
        int rl  = wave*32 + mt*16 + half*8 + j;        // C/D layout: M = j + 8*half
        int col = t*16 + l15;                          // N = l15
        float d = xx_lds[rl] + ww_lds[col] - 2.0f * acc[mt][t][j];
        float s = __expf(-ga_lds[col] * d) * al_lds[col];
        s_lds[col * S_PITCH + rl] = s;
      }
    }
  }
  __syncthreads();

  // ---------------- Dempster scan: one batch row per thread -----------------
  const int row = tid;
  float mx = 0.f;                                      // s > 0 always
  #pragma unroll 8
  for (int p = 0; p < P_DIM; ++p) mx = fmaxf(mx, s_lds[p*S_PITCH + row]);
  const float inv_mx = 1.f / (mx + EPS_F);             // si = s * inv_mx

  float m[C_DIM + 1];                                  // register-resident mass
  {
    float si0 = s_lds[row] * inv_mx;
    #pragma unroll
    for (int c = 0; c < C_DIM; ++c) m[c] = si0 * u_lds[c];  // uniform -> broadcast
    m[C_DIM] = 1.f - si0;                              // omega
  }

  #pragma unroll 1
  for (int p = 1; p < P_DIM; ++p) {
    const float sp  = s_lds[p*S_PITCH + row] * inv_mx;
    const float w2  = 1.f - sp;                        // omega2
    const float om1 = m[C_DIM];                        // omega1
    const float* up = u_lds + p * C_DIM;               // wave-uniform address
    float ss[4] = {0.f, 0.f, 0.f, 0.f};
    #pragma unroll
    for (int c = 0; c < C_DIM; ++c) {
      float m2v = sp * up[c];
      float v   = m[c] * (m2v + w2) + om1 * m2v;       // m1*m2 + m1*w2 + w1*m2
      m[c] = v;
      ss[c & 3] += v;
    }
    float vl = 3.f * om1 * w2;                         // elementwise rule on omega slot
    m[C_DIM] = vl;
    float sum = (ss[0] + ss[1]) + (ss[2] + ss[3]) + vl;
    float inv = 1.f / (sum + EPS_F);
    #pragma unroll
    for (int c = 0; c <= C_DIM; ++c) m[c] *= inv;
  }

  // final normalize (no EPS, per reference) and store
  float ts[4] = {0.f, 0.f, 0.f, 0.f};
  #pragma unroll
  for (int c = 0; c <= C_DIM; ++c) ts[c & 3] += m[c];
  const float invt = 1.f / ((ts[0] + ts[1]) + (ts[2] + ts[3]));
  float* op = out + (size_t)(row_base + row) * (C_DIM + 1);
  #pragma unroll 4
  for (int c = 0; c <= C_DIM; ++c) op[c] = m[c] * invt;
}

extern "C" void kernel_launch(void* const* d_in, const int* in_sizes, int n_in,
                              void* d_out, int out_size, void* d_ws, size_t ws_size,
                              hipStream_t stream) {
  (void)in_sizes; (void)n_in; (void)out_size; (void)d_ws; (void)ws_size;
  const float* x    = (const float*)d_in[0];
  const float* w    = (const float*)d_in[1];
  const float* xi   = (const float*)d_in[2];
  const float* eta  = (const float*)d_in[3];
  const float* beta = (const float*)d_in[4];
  float* out = (float*)d_out;
  dim3 grid(B_TOTAL / ROWS_PER_BLOCK);   // 128 blocks
  ds_fused<<<grid, NTHREADS, 0, stream>>>(x, w, xi, eta, beta, out);
}